// GraphProp_15083925143987
// MI455X (gfx1250) — compile-verified
//
#include <hip/hip_runtime.h>
#include <stdint.h>

#define N_NODES 50000
#define N_EDGES 800000
#define HID 64
#define H2  128
#define H3  192
#define ROUNDS 2

typedef float v2f __attribute__((ext_vector_type(2)));
typedef float v8f __attribute__((ext_vector_type(8)));

// D = A(16x4 f32) * B(4x16 f32) + C(16x16 f32), wave32 WMMA
__device__ __forceinline__ v8f wmma_f32(v2f a, v2f b, v8f c) {
    return __builtin_amdgcn_wmma_f32_16x16x4_f32(
        /*neg_a=*/false, a, /*neg_b=*/false, b,
        /*c_mod=*/(short)0, c, /*reuse_a=*/false, /*reuse_b=*/false);
}

__device__ __forceinline__ void atomic_add_f32(float* p, float v) {
    __hip_atomic_fetch_add(p, v, __ATOMIC_RELAXED, __HIP_MEMORY_SCOPE_AGENT);
}

__device__ __forceinline__ float sigmoidf(float x) {
    return 1.0f / (1.0f + __expf(-x));
}

__device__ __forceinline__ v8f splat8(float x) {
    v8f r;
#pragma unroll
    for (int i = 0; i < 8; ++i) r[i] = x;
    return r;
}

__global__ void zero_kernel(float* __restrict__ p, int n) {
    int i = blockIdx.x * blockDim.x + threadIdx.x;
    if (i < n) p[i] = 0.0f;
}

// One wave handles 16 edges: e_act[16,128] = concat(hv[dst],hv[src],he) @ w_msg^T + b
// then fused segment-sum via global float atomics into a_out[N,128].
__global__ __launch_bounds__(128) void edge_msg_kernel(
    const float* __restrict__ hv, const float* __restrict__ he,
    const int* __restrict__ src, const int* __restrict__ dst,
    const float* __restrict__ w_msg, const float* __restrict__ b_msg,
    float* __restrict__ a_out) {
    __shared__ float sm_all[4 * 16 * H3];           // 48 KB: 16x192 tile per wave
    const int lane = threadIdx.x & 31;
    const int wid  = threadIdx.x >> 5;
    const int grp  = blockIdx.x * 4 + wid;
    if (grp >= N_EDGES / 16) return;                // wave-uniform: EXEC stays full
    const int e0   = grp * 16;
    const int r16  = lane & 15;
    const int half = lane >> 4;
    float* sm = sm_all + wid * (16 * H3);

    // Stage msg_in tile [16 rows x 192] into LDS (coalesced 128-bit loads;
    // concat segments are 64-float aligned so each float4 stays in one source).
#pragma unroll 4
    for (int it = 0; it < 24; ++it) {
        int idx = it * 32 + lane;                   // 0..767 float4 chunks
        int row = idx / 48;
        int k   = (idx % 48) * 4;
        const float* p;
        if (k < 64)       p = hv + (int64_t)dst[e0 + row] * HID + k;
        else if (k < 128) p = hv + (int64_t)src[e0 + row] * HID + (k - 64);
        else              p = he + (int64_t)(e0 + row) * HID + (k - 128);
        *(float4*)(sm + row * H3 + k) = *(const float4*)p;
    }

    // Destination node of each of this lane's 8 C-matrix rows (M = j + 8*half)
    int dnode[8];
#pragma unroll
    for (int j = 0; j < 8; ++j) dnode[j] = dst[e0 + j + half * 8];

    const float* smrow = sm + r16 * H3;             // A row for this lane

#pragma unroll 1
    for (int nt = 0; nt < 8; ++nt) {                // 8 tiles cover N=128
        const int n = nt * 16 + r16;                // this lane's output column
        v8f acc = splat8(b_msg[n]);                 // bias is per-column
        const float* wrow = w_msg + (int64_t)n * H3;  // w_msg[n, :] (L2-resident)
#pragma unroll 4
        for (int s = 0; s < 48; ++s) {              // K = 192 in steps of 4
            int kb = 4 * s + half * 2;
            acc = wmma_f32(*(const v2f*)(smrow + kb),
                           *(const v2f*)(wrow + kb), acc);
        }
#pragma unroll
        for (int j = 0; j < 8; ++j)                 // fused segment-sum
            atomic_add_f32(a_out + (int64_t)dnode[j] * H2 + n, acc[j]);
    }
}

// One wave handles 16 nodes: full GRUCell with WMMA for gi = a@w_ih^T+b_ih and
// gh = h@w_hh^T+b_hh, gates in VALU/TRANS (co-executes with the matrix pipe).
__global__ __launch_bounds__(128) void gru_kernel(
    const float* __restrict__ a_in, const float* __restrict__ hv_cur,
    const float* __restrict__ w_ih, const float* __restrict__ w_hh,
    const float* __restrict__ b_ih, const float* __restrict__ b_hh,
    float* __restrict__ hv_next) {
    const int lane = threadIdx.x & 31;
    const int grp  = blockIdx.x * 4 + (threadIdx.x >> 5);
    if (grp >= N_NODES / 16) return;                // wave-uniform guard
    const int node0 = grp * 16;
    const int r16   = lane & 15;
    const int half  = lane >> 4;
    const float* arow = a_in  + (int64_t)(node0 + r16) * H2;
    const float* hrow = hv_cur + (int64_t)(node0 + r16) * HID;

#pragma unroll 1
    for (int c = 0; c < 4; ++c) {                   // hidden dim in 16-wide chunks
        const int hcol = c * 16 + r16;
        v8f giR = splat8(b_ih[hcol]);
        v8f giZ = splat8(b_ih[HID + hcol]);
        v8f giN = splat8(b_ih[2 * HID + hcol]);
        v8f ghR = splat8(b_hh[hcol]);
        v8f ghZ = splat8(b_hh[HID + hcol]);
        v8f ghN = splat8(b_hh[2 * HID + hcol]);
        const float* wiR = w_ih + (int64_t)(hcol) * H2;
        const float* wiZ = w_ih + (int64_t)(HID + hcol) * H2;
        const float* wiN = w_ih + (int64_t)(2 * HID + hcol) * H2;
        const float* whR = w_hh + (int64_t)(hcol) * HID;
        const float* whZ = w_hh + (int64_t)(HID + hcol) * HID;
        const float* whN = w_hh + (int64_t)(2 * HID + hcol) * HID;

#pragma unroll 4
        for (int s = 0; s < 32; ++s) {              // K = 128 over a
            int kb = 4 * s + half * 2;
            v2f av = *(const v2f*)(arow + kb);
            giR = wmma_f32(av, *(const v2f*)(wiR + kb), giR);
            giZ = wmma_f32(av, *(const v2f*)(wiZ + kb), giZ);
            giN = wmma_f32(av, *(const v2f*)(wiN + kb), giN);
        }
#pragma unroll 4
        for (int s = 0; s < 16; ++s) {              // K = 64 over hv
            int kb = 4 * s + half * 2;
            v2f hh = *(const v2f*)(hrow + kb);
            ghR = wmma_f32(hh, *(const v2f*)(whR + kb), ghR);
            ghZ = wmma_f32(hh, *(const v2f*)(whZ + kb), ghZ);
            ghN = wmma_f32(hh, *(const v2f*)(whN + kb), ghN);
        }
#pragma unroll
        for (int j = 0; j < 8; ++j) {               // gate math per C element
            int m = node0 + j + half * 8;
            float hprev = hv_cur[(int64_t)m * HID + c * 16 + r16];
            float r  = sigmoidf(giR[j] + ghR[j]);
            float z  = sigmoidf(giZ[j] + ghZ[j]);
            float nn = tanhf(giN[j] + r * ghN[j]);
            hv_next[(int64_t)m * HID + c * 16 + r16] = (1.0f - z) * nn + z * hprev;
        }
    }
}

extern "C" void kernel_launch(void* const* d_in, const int* in_sizes, int n_in,
                              void* d_out, int out_size, void* d_ws, size_t ws_size,
                              hipStream_t stream) {
    const float* hv0   = (const float*)d_in[0];
    const float* he    = (const float*)d_in[1];
    const int*   src   = (const int*)d_in[2];
    const int*   dst   = (const int*)d_in[3];
    const float* w_msg = (const float*)d_in[4];
    const float* b_msg = (const float*)d_in[5];
    const float* w_ih  = (const float*)d_in[6];
    const float* w_hh  = (const float*)d_in[7];
    const float* b_ih  = (const float*)d_in[8];
    const float* b_hh  = (const float*)d_in[9];

    float* a_buf  = (float*)d_ws;                         // N*128 floats (25.6 MB)
    float* hv_buf = a_buf + (size_t)N_NODES * H2;         // N*64  floats (12.8 MB)
    float* buf0 = (float*)d_out;                          // round-even hv lives here
    float* buf1 = hv_buf;

    hipMemcpyAsync(buf0, hv0, (size_t)N_NODES * HID * sizeof(float),
                   hipMemcpyDeviceToDevice, stream);

    for (int t = 0; t < ROUNDS; ++t) {
        float* cur = (t & 1) ? buf1 : buf0;
        float* nxt = (t & 1) ? buf0 : buf1;
        const int nz = N_NODES * H2;
        zero_kernel<<<(nz + 255) / 256, 256, 0, stream>>>(a_buf, nz);
        edge_msg_kernel<<<(N_EDGES / 16) / 4, 128, 0, stream>>>(
            cur, he, src, dst,
            w_msg + (size_t)t * H2 * H3, b_msg + (size_t)t * H2, a_buf);
        gru_kernel<<<((N_NODES / 16) + 3) / 4, 128, 0, stream>>>(
            a_buf, cur,
            w_ih + (size_t)t * H3 * H2, w_hh + (size_t)t * H3 * HID,
            b_ih + (size_t)t * H3, b_hh + (size_t)t * H3, nxt);
    }
    // ROUNDS is even -> final hv ends in buf0 == d_out
}